// ActionModel_80092550136175
// MI455X (gfx1250) — compile-verified
//
#include <hip/hip_runtime.h>

// Problem constants (match reference)
#define B_    32
#define L_    512
#define D_    1024
#define KLIN  7
#define NTOK  (B_ * L_)      // 16384 tokens

#define KC   64              // K-chunk staged in LDS
#define STR  68              // padded row stride (floats): +4 pad -> conflict-free banks

typedef __attribute__((ext_vector_type(2))) float v2f;
typedef __attribute__((ext_vector_type(4))) float v4f;
typedef __attribute__((ext_vector_type(8))) float v8f;
typedef __attribute__((ext_vector_type(4))) int   v4i;

// -------- optional CDNA5 async global->LDS path (ASYNCcnt, no VGPR roundtrip)
#ifndef USE_ASYNC_LDS
#define USE_ASYNC_LDS 1
#endif
#if USE_ASYNC_LDS && defined(__has_builtin)
#  if __has_builtin(__builtin_amdgcn_global_load_async_to_lds_b128)
#    define HAVE_ASYNC_B128 1
#  endif
#endif
#ifndef HAVE_ASYNC_B128
#  define HAVE_ASYNC_B128 0
#endif

#if HAVE_ASYNC_B128
// Builtin signature (from hipcc diagnostic): (v4i AS1* src, v4i AS3* dst, imm, imm)
typedef __attribute__((address_space(1))) v4i as1_v4i;
typedef __attribute__((address_space(3))) v4i as3_v4i;
__device__ __forceinline__ void async_copy16(const float* g, float* l) {
    // 16 bytes per lane, memory -> LDS, tracked by ASYNCcnt
    __builtin_amdgcn_global_load_async_to_lds_b128(
        (as1_v4i*)g, (as3_v4i*)l, /*offset=*/0, /*cpol=*/0);
}
__device__ __forceinline__ void wait_async0() {
    asm volatile("s_wait_asynccnt 0x0" ::: "memory");
}
#endif

// ---------------------------------------------------------------- routing ---

__global__ void init_counts_kernel(int* __restrict__ counts) {
    if (threadIdx.x < 8) counts[threadIdx.x] = 0;
}

__global__ void route_tokens_kernel(const int* __restrict__ atype,
                                    int* __restrict__ counts,
                                    int* __restrict__ idx) {
    int t = blockIdx.x * blockDim.x + threadIdx.x;
    if (t >= NTOK) return;
    int ty = atype[t];
    if (ty > 0) {
        int e = ty - 1;                       // expert 0..6
        int pos = atomicAdd(&counts[e], 1);
        idx[e * NTOK + pos] = t;
    }
}

// Identity ('root') expert: straight row copy for action_type == 0.
__global__ __launch_bounds__(128)
void copy_identity_kernel(const float* __restrict__ actions,
                          const int* __restrict__ atype,
                          float* __restrict__ out) {
    int tok = blockIdx.x;
    if (atype[tok] != 0) return;
    const v4f* src = (const v4f*)(actions + (size_t)tok * D_);
    v4f*       dst = (v4f*)(out + (size_t)tok * D_);
    dst[threadIdx.x]       = src[threadIdx.x];
    dst[threadIdx.x + 128] = src[threadIdx.x + 128];
}

// ---------------------------------------------------- grouped GEMM (WMMA) ---
// Block tile: 16 (M = gathered tokens) x 256 (N). 256 threads = 8 wave32;
// wave w owns N rows [bx*256 + w*32, +32): two 16x16 accumulators c0,c1 so
// each A fetch feeds two v_wmma_f32_16x16x4_f32.

__global__ __launch_bounds__(256)
void grouped_gemm_kernel(const float* __restrict__ actions,
                         const float* __restrict__ W,
                         const float* __restrict__ bias,
                         const int* __restrict__ counts,
                         const int* __restrict__ idx,
                         float* __restrict__ out) {
    __shared__ float Xs[16 * STR];            //  4.25 KB
    __shared__ float Ws[8 * 32 * STR];        // 68    KB (32 rows per wave)

    const int e      = blockIdx.z;
    const int cnt    = counts[e];
    const int m_base = blockIdx.y * 16;
    if (m_base >= cnt) return;                // uniform early-exit

    const int wave   = threadIdx.x >> 5;
    const int lane   = threadIdx.x & 31;
    const int nl     = lane & 15;             // col within 16-wide tile
    const int khalf  = (lane < 16) ? 0 : 2;   // K pair selected by lane half
    const int n_wave = blockIdx.x * 256 + wave * 32;

    const int*   my_idx = idx + e * NTOK;
    const float* We     = W + (size_t)e * D_ * D_;
    float*       Wr     = &Ws[wave * 32 * STR];

    // X staging slot for this thread: one float4 of row xrow
    const int xrow = threadIdx.x >> 4;        // 0..15
    const int xc4  = threadIdx.x & 15;        // 16 float4 per KC row
    // W staging: rows 2r + (lane>>4), float4 index lane&15
    const int wrow_off = lane >> 4;
    const int wc4      = lane & 15;

    const bool full = (cnt - m_base) >= 16;   // wave-uniform

    v8f c0 = {}, c1 = {};

    for (int k0 = 0; k0 < D_; k0 += KC) {
        __syncthreads();                      // protect LDS reuse across chunks

#if HAVE_ASYNC_B128
        if (full) {
            // ---- async stage X tile (1 op/thread) and W slice (16 ops/thread)
            int tok = my_idx[m_base + xrow];
            async_copy16(actions + (size_t)tok * D_ + k0 + xc4 * 4,
                         &Xs[xrow * STR + xc4 * 4]);
            #pragma unroll
            for (int r = 0; r < 16; ++r) {
                int row = 2 * r + wrow_off;
                async_copy16(We + (size_t)(n_wave + row) * D_ + k0 + wc4 * 4,
                             &Wr[row * STR + wc4 * 4]);
            }
            wait_async0();                    // this wave's asyncs visible in LDS
        } else
#endif
        {
            // ---- manual stage, zero-padding tail rows of X
            v4f vx = {};
            int mrow = m_base + xrow;
            if (mrow < cnt) {
                int tok = my_idx[mrow];
                vx = *(const v4f*)(actions + (size_t)tok * D_ + k0 + xc4 * 4);
            }
            *(v4f*)(&Xs[xrow * STR + xc4 * 4]) = vx;
            #pragma unroll
            for (int r = 0; r < 16; ++r) {
                int row = 2 * r + wrow_off;
                v4f vw = *(const v4f*)(We + (size_t)(n_wave + row) * D_ + k0 + wc4 * 4);
                *(v4f*)(&Wr[row * STR + wc4 * 4]) = vw;
            }
        }

        __syncthreads();

        // ---- 32 WMMAs per chunk: one A fetch feeds two independent tiles
        #pragma unroll
        for (int kk = 0; kk < KC; kk += 4) {
            v2f a  = *(const v2f*)(&Xs[nl * STR + kk + khalf]);
            v2f b0 = *(const v2f*)(&Wr[nl * STR + kk + khalf]);
            v2f b1 = *(const v2f*)(&Wr[(16 + nl) * STR + kk + khalf]);
            c0 = __builtin_amdgcn_wmma_f32_16x16x4_f32(
                     false, a, false, b0, (short)0, c0, false, false);
            c1 = __builtin_amdgcn_wmma_f32_16x16x4_f32(
                     false, a, false, b1, (short)0, c1, false, false);
        }
    }

    // ---- bias (depends only on N), then scatter both tiles
    float bv0 = bias[e * D_ + n_wave + nl];
    float bv1 = bias[e * D_ + n_wave + 16 + nl];
    #pragma unroll
    for (int i = 0; i < 8; ++i) { c0[i] += bv0; c1[i] += bv1; }

    #pragma unroll
    for (int i = 0; i < 8; ++i) {
        int m    = (lane < 16) ? i : (i + 8);
        int mrow = m_base + m;
        if (mrow < cnt) {
            int tok = my_idx[mrow];
            float* dst = out + (size_t)tok * D_ + n_wave + nl;
            dst[0]  = c0[i];
            dst[16] = c1[i];
        }
    }
}

// ------------------------------------------------------------------ launch ---

extern "C" void kernel_launch(void* const* d_in, const int* in_sizes, int n_in,
                              void* d_out, int out_size, void* d_ws, size_t ws_size,
                              hipStream_t stream) {
    const float* actions = (const float*)d_in[0];   // (B, L, D) fp32
    const int*   atype   = (const int*)  d_in[1];   // (B, L) int32 in [0, 8)
    const float* W       = (const float*)d_in[2];   // (7, D, D) fp32  (out, in)
    const float* bias    = (const float*)d_in[3];   // (7, D) fp32
    float*       out     = (float*)d_out;           // (B, L, D) fp32

    // Workspace: [0,32)=counts (8 ints), [64,...) = 7 index lists of NTOK ints.
    int* counts = (int*)d_ws;
    int* idx    = (int*)((char*)d_ws + 64);

    init_counts_kernel<<<1, 32, 0, stream>>>(counts);
    route_tokens_kernel<<<(NTOK + 255) / 256, 256, 0, stream>>>(atype, counts, idx);
    copy_identity_kernel<<<NTOK, 128, 0, stream>>>(actions, atype, out);

    dim3 grid(D_ / 256, NTOK / 16, KLIN);   // (4 N-tiles, 1024 M-tiles, 7 experts)
    grouped_gemm_kernel<<<grid, 256, 0, stream>>>(actions, W, bias, counts, idx, out);
}